// KMeansBaseline_80015240725022
// MI455X (gfx1250) — compile-verified
//
#include <hip/hip_runtime.h>

typedef float v2f __attribute__((ext_vector_type(2)));
typedef float v8f __attribute__((ext_vector_type(8)));

#define K_CLUSTERS 64
#define N_ITERS 20
#define TPB 512

// One workgroup per batch. Centroids + accumulators live in LDS.
//
// Scoring tile (16 centroids x 16 points) via V_WMMA_F32_16X16X4_F32:
//   A row m (centroid): (-2cx, -2cy, |c|^2, 0)
//   B col n (point):    (x, y, 1, 0)
//   D[m][n] = -2*(c.p) + |c|^2  -> argmin_m D == argmin_m |p - c|^2
//
// D layout (32-bit C/D 16x16): VGPR j, lanes 0-15 -> (m=j, n=lane),
// lanes 16-31 -> (m=j+8, n=lane-16): point n's 64 scores live in lanes
// n and n+16 only.
//
// Argmin: embed the 6-bit cluster id in the low mantissa bits of the f32
// score; numeric min of the keyed scores == argmin. 31-op fmin tree
// (depth 5) + one xor-16 exchange instead of a 64-deep cndmask chain.
__global__ __launch_bounds__(TPB) void kmeans_wmma_kernel(
    const float* __restrict__ coords,   // (B, N, 2)
    const int*   __restrict__ init_idx, // (B, 64)
    float*       __restrict__ out,      // (B, N) assignments
    int nPoints)
{
    __shared__ float cx[K_CLUSTERS], cy[K_CLUSTERS];
    __shared__ float sx[K_CLUSTERS], sy[K_CLUSTERS], sc[K_CLUSTERS];

    const int b      = blockIdx.x;
    const int t      = threadIdx.x;
    const int lane   = t & 31;          // wave32
    const int wave   = t >> 5;
    const int nWaves = TPB / 32;
    const int nTiles = nPoints / 16;    // 20000/16 = 1250, exact

    const float*  pts  = coords + (size_t)b * nPoints * 2;
    const float2* pts2 = reinterpret_cast<const float2*>(pts);
    float*        outb = out + (size_t)b * nPoints;

    if (t < K_CLUSTERS) {
        int idx = init_idx[b * K_CLUSTERS + t];
        cx[t] = pts[2 * idx + 0];
        cy[t] = pts[2 * idx + 1];
    }

    const int  nsub  = lane & 15;
    const bool hi    = (lane >= 16);
    const int  hbase = hi ? 8 : 0;      // centroid-row offset carried by this half

    // ---- 19 accumulate + update iterations (centroids_0 .. centroids_19) ----
    for (int iter = 0; iter < N_ITERS - 1; ++iter) {
        if (t < K_CLUSTERS) { sx[t] = 0.f; sy[t] = 0.f; sc[t] = 0.f; }
        __syncthreads();   // accumulators zeroed; centroids stable

        // A fragments (centroids), 4 tiles of 16 rows.
        v2f af[4];
#pragma unroll
        for (int ct = 0; ct < 4; ++ct) {
            float x = cx[ct * 16 + nsub];
            float y = cy[ct * 16 + nsub];
            af[ct].x = hi ? (x * x + y * y) : (-2.f * x);
            af[ct].y = hi ? 0.f             : (-2.f * y);
        }

        for (int tile = wave; tile < nTiles; tile += nWaves) {
            const int base = tile * 16;
            __builtin_prefetch(&pts2[base + nWaves * 16], 0, 3);

            float2 p = pts2[base + nsub];          // uniform load, hi half redundant
            v2f bfrag;
            bfrag.x = hi ? 1.f : p.x;              // rows K0/K2
            bfrag.y = hi ? 0.f : p.y;              // rows K1/K3

            float key[32];
#pragma unroll
            for (int ct = 0; ct < 4; ++ct) {
                v8f c = {};
                v8f d = __builtin_amdgcn_wmma_f32_16x16x4_f32(
                    false, af[ct], false, bfrag, (short)0, c, false, false);
#pragma unroll
                for (int j = 0; j < 8; ++j) {
                    int bits = __float_as_int(d[j]);
                    key[ct * 8 + j] =
                        __int_as_float((bits & ~63) | (ct * 16 + j) | hbase);
                }
            }
            // in-lane min tree (depth 5), then combine the two halves
#pragma unroll
            for (int s = 16; s >= 1; s >>= 1)
#pragma unroll
                for (int k = 0; k < s; ++k) key[k] = fminf(key[k], key[k + s]);
            float kx = __shfl_xor(key[0], 16, 32);
            float k0 = fminf(key[0], kx);
            int bestI = __float_as_int(k0) & 63;

            if (!hi) {                              // lane n owns point base+n
                atomicAdd(&sx[bestI], p.x);         // ds_add_f32
                atomicAdd(&sy[bestI], p.y);
                atomicAdd(&sc[bestI], 1.f);
            }
        }
        __syncthreads();   // all accumulation done

        if (t < K_CLUSTERS) {
            float c = sc[t];
            if (c > 0.f) { cx[t] = sx[t] / c; cy[t] = sy[t] / c; }
        }
        __syncthreads();   // updated centroids visible
    }

    // ---- final assignment pass (ids recorded with centroids after 19 updates) ----
    {
        v2f af[4];
#pragma unroll
        for (int ct = 0; ct < 4; ++ct) {
            float x = cx[ct * 16 + nsub];
            float y = cy[ct * 16 + nsub];
            af[ct].x = hi ? (x * x + y * y) : (-2.f * x);
            af[ct].y = hi ? 0.f             : (-2.f * y);
        }

        for (int tile = wave; tile < nTiles; tile += nWaves) {
            const int base = tile * 16;
            __builtin_prefetch(&pts2[base + nWaves * 16], 0, 3);

            float2 p = pts2[base + nsub];
            v2f bfrag;
            bfrag.x = hi ? 1.f : p.x;
            bfrag.y = hi ? 0.f : p.y;

            float key[32];
#pragma unroll
            for (int ct = 0; ct < 4; ++ct) {
                v8f c = {};
                v8f d = __builtin_amdgcn_wmma_f32_16x16x4_f32(
                    false, af[ct], false, bfrag, (short)0, c, false, false);
#pragma unroll
                for (int j = 0; j < 8; ++j) {
                    int bits = __float_as_int(d[j]);
                    key[ct * 8 + j] =
                        __int_as_float((bits & ~63) | (ct * 16 + j) | hbase);
                }
            }
#pragma unroll
            for (int s = 16; s >= 1; s >>= 1)
#pragma unroll
                for (int k = 0; k < s; ++k) key[k] = fminf(key[k], key[k + s]);
            float kx = __shfl_xor(key[0], 16, 32);
            float k0 = fminf(key[0], kx);
            int bestI = __float_as_int(k0) & 63;

            if (!hi) outb[base + lane] = (float)bestI;
        }
    }
}

extern "C" void kernel_launch(void* const* d_in, const int* in_sizes, int n_in,
                              void* d_out, int out_size, void* d_ws, size_t ws_size,
                              hipStream_t stream) {
    const float* coords   = (const float*)d_in[0];   // (B, N, 2) float32
    const int*   init_idx = (const int*)d_in[1];     // (B, 64) int32
    float*       out      = (float*)d_out;           // (B, N) assignments

    const int B = in_sizes[1] / K_CLUSTERS;          // 32
    const int N = in_sizes[0] / (B * 2);             // 20000

    kmeans_wmma_kernel<<<B, TPB, 0, stream>>>(coords, init_idx, out, N);
}